// EmbeddingLoss_91311004713287
// MI455X (gfx1250) — compile-verified
//
#include <hip/hip_runtime.h>
#include <math.h>

// ---------------------------------------------------------------------------
// EmbeddingLoss for MI455X (gfx1250, wave32).
//
//   k_zero   : zero workspace accumulators
//   k_accum  : per-(b,s,c) sums of {count, margin, emb[8]} via LDS atomics
//   k_stats  : centroids, sigma, cent^2, 1/(2s^2+eps), present, n_sub
//   k_main   : subset-specialized pass; dists via chained
//              V_WMMA_F32_16X16X4_F32 (D = (-2*cent) x embT + (cent2+e2)),
//              then BCE = -log(oh ? pc : 1-pc) (exact algebraic collapse of
//              the reference's stable_bce(log(pc/(1-pc)), oh))
//   k_final  : present-masked class/subset means -> scalar loss
// ---------------------------------------------------------------------------

#define NCLS 16
#define NSUB 16           // 4 batches x 4 used semantic labels (s==4 dropped)
#define NGRP 256          // NSUB * NCLS

typedef float v2f __attribute__((ext_vector_type(2)));
typedef float v8f __attribute__((ext_vector_type(8)));

// workspace layout (float offsets)
#define WS_CNT   0        // [256]   per-group counts
#define WS_MSUM  256      // [256]   per-group margin sums
#define WS_ESUM  512      // [256*8] per-group embedding sums
#define WS_BCE   2560     // [256]   per-group bce sums
#define WS_SM    2816     // [256]   per-group smooth sums
#define WS_A2C   3072     // [256*8] -2*centroid (WMMA A operand source)
#define WS_C2    5120     // [256]   |centroid|^2
#define WS_SIG   5376     // [256]   sigma
#define WS_INVD  5632     // [256]   1/(2*sigma^2+1e-8)
#define WS_PRES  5888     // [256]   present flag
#define WS_NSUB  6144     // [16]    subset sizes (clamped >=1)
#define WS_TOT   6160

__global__ void k_zero(float* ws, int n) {
    int i = blockIdx.x * blockDim.x + threadIdx.x;
    if (i < n) ws[i] = 0.0f;
}

__global__ __launch_bounds__(256)
void k_accum(const float* __restrict__ emb, const float* __restrict__ marg,
             const int* __restrict__ slab, const int* __restrict__ clab,
             const int* __restrict__ bidx, float* __restrict__ ws, int N) {
    __shared__ float lcnt[NGRP];
    __shared__ float lm[NGRP];
    __shared__ float le[NGRP * 8];
    for (int i = threadIdx.x; i < NGRP; i += blockDim.x) { lcnt[i] = 0.f; lm[i] = 0.f; }
    for (int i = threadIdx.x; i < NGRP * 8; i += blockDim.x) le[i] = 0.f;
    __syncthreads();

    for (long n = (long)blockIdx.x * blockDim.x + threadIdx.x; n < N;
         n += (long)gridDim.x * blockDim.x) {
        int s = slab[n];
        if (s >= 4) continue;                      // s==4 excluded by reference
        int g = (bidx[n] * 4 + s) * 16 + clab[n];
        atomicAdd(&lcnt[g], 1.0f);
        atomicAdd(&lm[g], marg[n]);
        const float* e = emb + n * 8;
        #pragma unroll
        for (int d = 0; d < 8; ++d) atomicAdd(&le[g * 8 + d], e[d]);
    }
    __syncthreads();
    for (int i = threadIdx.x; i < NGRP; i += blockDim.x) {
        atomicAdd(&ws[WS_CNT + i], lcnt[i]);
        atomicAdd(&ws[WS_MSUM + i], lm[i]);
    }
    for (int i = threadIdx.x; i < NGRP * 8; i += blockDim.x)
        atomicAdd(&ws[WS_ESUM + i], le[i]);
}

__global__ void k_stats(float* ws) {
    int g = threadIdx.x;                           // 256 threads, one per group
    float cnt  = ws[WS_CNT + g];
    float pres = (cnt > 0.f) ? 1.f : 0.f;
    float cdiv = (cnt > 0.f) ? cnt : 1.f;
    float c2 = 0.f;
    #pragma unroll
    for (int d = 0; d < 8; ++d) {
        float cen = ws[WS_ESUM + g * 8 + d] / cdiv;
        ws[WS_A2C + g * 8 + d] = -2.0f * cen;      // pre-scaled A operand
        c2 += cen * cen;
    }
    float sig = ws[WS_MSUM + g] / cdiv;
    ws[WS_C2 + g]   = c2;
    ws[WS_SIG + g]  = sig;
    ws[WS_INVD + g] = 1.0f / (2.0f * sig * sig + 1e-8f);
    ws[WS_PRES + g] = pres;
    __syncthreads();
    if (g < NSUB) {                                // subset sizes
        float s = 0.f;
        for (int c = 0; c < NCLS; ++c) s += ws[WS_CNT + g * 16 + c];
        ws[WS_NSUB + g] = fmaxf(s, 1.0f);
    }
}

__global__ __launch_bounds__(256)
void k_main(const float* __restrict__ emb, const float* __restrict__ marg,
            const int* __restrict__ slab, const int* __restrict__ clab,
            const int* __restrict__ bidx, float* __restrict__ ws, int N) {
    const int sub   = blockIdx.y;                  // one subset per block-row
    const int b_sub = sub >> 2;
    const int s_sub = sub & 3;
    const int gbase = sub * NCLS;

    __shared__ float lds_bce[NCLS];
    __shared__ float lds_sm[NCLS];
    if (threadIdx.x < NCLS) { lds_bce[threadIdx.x] = 0.f; lds_sm[threadIdx.x] = 0.f; }
    __syncthreads();

    const int lane = threadIdx.x & 31;
    const int wv   = threadIdx.x >> 5;
    const int hh   = lane >> 4;                    // lane half (f32 WMMA layout)
    const bool hi  = (hh != 0);
    const int n16  = lane & 15;

    // A operand (-2*cent), 16x4 f32 layout: V0 = K(2*hh), V1 = K(2*hh+1), M = lane%16
    const float* a2 = ws + WS_A2C + (gbase + n16) * 8;
    v2f A0, A1;
    A0.x = a2[2 * hh];     A0.y = a2[2 * hh + 1];
    A1.x = a2[4 + 2 * hh]; A1.y = a2[5 + 2 * hh];

    float cc[8], ninvd[8];
    #pragma unroll
    for (int r = 0; r < 8; ++r) {                  // class c = r + 8*hh
        cc[r]    = ws[WS_C2 + gbase + r + 8 * hh];
        ninvd[r] = -ws[WS_INVD + gbase + r + 8 * hh];
    }
    float acc[8];
    #pragma unroll
    for (int r = 0; r < 8; ++r) acc[r] = 0.f;

    const int  wavesPerBlk = blockDim.x >> 5;
    const long stride = (long)gridDim.x * wavesPerBlk * 16;
    // wave-uniform loop bounds => EXEC all-ones around WMMA
    for (long base = ((long)blockIdx.x * wavesPerBlk + wv) * 16; base < N; base += stride) {
        long idx = base + n16;
        bool inr = idx < N;
        long ii  = inr ? idx : 0;
        const float* e = emb + ii * 8;
        float4 eLo = *(const float4*)(e);
        float4 eHi = *(const float4*)(e + 4);
        if (base + stride + 16 <= N)
            __builtin_prefetch(emb + (base + stride) * 8, 0, 0);   // global_prefetch_b8
        // unconditional (clamped-index) label loads: keep loop branch-free
        int cl = clab[ii];
        int bi = bidx[ii];
        int sl = slab[ii];
        bool match = inr & (bi == b_sub) & (sl == s_sub);
        float msk  = match ? 1.f : 0.f;

        float e2 = eLo.x*eLo.x + eLo.y*eLo.y + eLo.z*eLo.z + eLo.w*eLo.w
                 + eHi.x*eHi.x + eHi.y*eHi.y + eHi.z*eHi.z + eHi.w*eHi.w;
        // B operand 4x16 f32: V0 = K(2*hh), V1 = K(2*hh+1), N = lane%16
        // hh in {0,1} -> single v_cndmask per component (no dynamic indexing)
        v2f B0, B1;
        B0.x = hi ? eLo.z : eLo.x;  B0.y = hi ? eLo.w : eLo.y;
        B1.x = hi ? eHi.z : eHi.x;  B1.y = hi ? eHi.w : eHi.y;

        // C seeded with cent2[c] + e2[n] so D is the distance matrix directly
        v8f c;
        #pragma unroll
        for (int r = 0; r < 8; ++r) c[r] = cc[r] + e2;
        c = __builtin_amdgcn_wmma_f32_16x16x4_f32(false, A0, false, B0, (short)0, c, false, false);
        c = __builtin_amdgcn_wmma_f32_16x16x4_f32(false, A1, false, B1, (short)0, c, false, false);

        int clr = cl - 8 * hh;                     // compare against unroll const r
        #pragma unroll
        for (int r = 0; r < 8; ++r) {
            float t  = c[r];                               // dists[c][n]
            // pc = clamp(exp(-t/(2s^2+eps)), 1e-6, 1-1e-6); upper clamp subsumes clip-to-1
            float pc = fminf(fmaxf(__expf(t * ninvd[r]), 1e-6f), 1.0f - 1e-6f);
            // stable_bce(log(pc/(1-pc)), oh) == -log(oh ? pc : 1-pc)  (exact identity)
            float sel = (clr == r) ? pc : 1.0f - pc;
            acc[r] += msk * (-__logf(sel));
        }
        if (match & (hh == 0)) {                           // each point counted once
            float dm = marg[ii] - ws[WS_SIG + gbase + cl];
            atomicAdd(&lds_sm[cl], dm * dm);
        }
    }

    // reduce acc over 16-lane halves: lane 0 -> class r, lane 16 -> class r+8
    #pragma unroll
    for (int r = 0; r < 8; ++r) {
        float v = acc[r];
        #pragma unroll
        for (int o = 1; o < 16; o <<= 1) v += __shfl_xor(v, o, 32);
        if (n16 == 0) atomicAdd(&lds_bce[r + 8 * hh], v);
    }
    __syncthreads();
    if (threadIdx.x < NCLS) {
        atomicAdd(&ws[WS_BCE + gbase + threadIdx.x], lds_bce[threadIdx.x]);
        atomicAdd(&ws[WS_SM + gbase + threadIdx.x], lds_sm[threadIdx.x]);
    }
}

__global__ void k_final(const float* __restrict__ ws, float* __restrict__ out) {
    __shared__ float lsum[NSUB];
    int t = threadIdx.x;
    if (t < NSUB) {
        float ns = ws[WS_NSUB + t];
        float sb = 0.f, ss = 0.f;
        int   np = 0;
        for (int c = 0; c < NCLS; ++c) {
            int g = t * NCLS + c;
            if (ws[WS_PRES + g] > 0.f) {
                np++;
                sb += ws[WS_BCE + g] / ns;
                ss += ws[WS_SM + g];
            }
        }
        float ncl = (np > 0) ? (float)np : 1.f;
        lsum[t] = sb / ncl + ss / ncl;        // EMB_W = SMOOTH_W = 1
    }
    __syncthreads();
    if (t == 0) {
        float s = 0.f;
        for (int i = 0; i < NSUB; ++i) s += lsum[i];
        *out = s / 16.0f;                     // mean over b of mean over s
    }
}

extern "C" void kernel_launch(void* const* d_in, const int* in_sizes, int n_in,
                              void* d_out, int out_size, void* d_ws, size_t ws_size,
                              hipStream_t stream) {
    const float* emb  = (const float*)d_in[0];
    const float* marg = (const float*)d_in[1];
    const int*   slab = (const int*)d_in[2];
    const int*   clab = (const int*)d_in[3];
    const int*   bidx = (const int*)d_in[4];
    int N = in_sizes[2];
    float* ws  = (float*)d_ws;
    float* out = (float*)d_out;

    k_zero<<<(WS_TOT + 255) / 256, 256, 0, stream>>>(ws, WS_TOT);
    k_accum<<<240, 256, 0, stream>>>(emb, marg, slab, clab, bidx, ws, N);
    k_stats<<<1, 256, 0, stream>>>(ws);
    dim3 g3(48, NSUB);
    k_main<<<g3, 256, 0, stream>>>(emb, marg, slab, clab, bidx, ws, N);
    k_final<<<1, 64, 0, stream>>>(ws, out);
}